// GraphConvsg_15925738733684
// MI455X (gfx1250) — compile-verified
//
#include <hip/hip_runtime.h>

typedef __attribute__((ext_vector_type(2))) float v2f;
typedef __attribute__((ext_vector_type(8))) float v8f;

// ---------------------------------------------------------------------------
// Kernel 1: zero workspace (deg + agg), vectorized float4 grid-stride
// ---------------------------------------------------------------------------
__global__ void zero_kernel(float4* __restrict__ p, long n4) {
    long i = (long)blockIdx.x * blockDim.x + threadIdx.x;
    long stride = (long)gridDim.x * blockDim.x;
    float4 z = make_float4(0.f, 0.f, 0.f, 0.f);
    for (; i < n4; i += stride) p[i] = z;
}

// ---------------------------------------------------------------------------
// Kernel 2: in-degree via float atomic add (exact for deg < 2^24)
// ---------------------------------------------------------------------------
__device__ __forceinline__ void atomic_add_f32(float* p, float v) {
    __hip_atomic_fetch_add(p, v, __ATOMIC_RELAXED, __HIP_MEMORY_SCOPE_AGENT);
}

__global__ void deg_kernel(const int* __restrict__ dst, float* __restrict__ deg,
                           int n_edges) {
    int e = blockIdx.x * blockDim.x + threadIdx.x;
    if (e < n_edges) atomic_add_f32(&deg[dst[e]], 1.0f);
}

// ---------------------------------------------------------------------------
// Kernel 3: norm = clip(deg,1)^-0.5, in place
// ---------------------------------------------------------------------------
__global__ void norm_kernel(float* __restrict__ deg, int n_nodes) {
    int i = blockIdx.x * blockDim.x + threadIdx.x;
    if (i < n_nodes) deg[i] = rsqrtf(fmaxf(deg[i], 1.0f));
}

// ---------------------------------------------------------------------------
// Kernel 4: edge scatter-add. One wave32 per edge; lane L owns features
// [4L..4L+3] -> float4 gather of feat[src], scale by norm[src], 4 fp32
// global atomics into agg[dst]. agg rows are L2-resident (51 MB < 192 MB).
// ---------------------------------------------------------------------------
__global__ void agg_kernel(const float* __restrict__ feat,
                           const int* __restrict__ src,
                           const int* __restrict__ dst,
                           const float* __restrict__ norm,
                           float* __restrict__ agg,
                           int n_edges) {
    int edge = (int)((blockIdx.x * (long)blockDim.x + threadIdx.x) >> 5);
    int lane = threadIdx.x & 31;
    if (edge >= n_edges) return;
    int s = src[edge];
    int d = dst[edge];
    float nrm = norm[s];
    const float4* frow = (const float4*)(feat + (size_t)s * 128);
    float4 v = frow[lane];                       // global_load_b128
    float* arow = agg + (size_t)d * 128 + lane * 4;
    atomic_add_f32(arow + 0, v.x * nrm);
    atomic_add_f32(arow + 1, v.y * nrm);
    atomic_add_f32(arow + 2, v.z * nrm);
    atomic_add_f32(arow + 3, v.w * nrm);
}

// ---------------------------------------------------------------------------
// Kernel 5: out = (agg @ W) * norm[row] + bias[col] using V_WMMA_F32_16X16X4_F32.
// Block = 256 threads = 8 waves. Block b owns rows [16b,16b+16); wave w owns
// cols [16w,16w+16). K=128 -> 32 chained f32 WMMAs, fp32 end-to-end.
//
// f32 WMMA VGPR layout (ISA 7.12.2):
//  A 16x4:  lanes 0-15: M=lane,  {v0,v1}={K,K+1}; lanes 16-31: M=lane-16, {K+2,K+3}
//  B 4x16:  lanes 0-15: N=lane,  {v0,v1}={K,K+1}; lanes 16-31: N=lane-16, {K+2,K+3}
//  C/D 16x16: VGPR r: lanes 0-15 -> M=r, N=lane; lanes 16-31 -> M=r+8, N=lane-16
// ---------------------------------------------------------------------------
__global__ void gemm_wmma_kernel(const float* __restrict__ agg,
                                 const float* __restrict__ weight, // [128][128]
                                 const float* __restrict__ bias,
                                 const float* __restrict__ norm,
                                 float* __restrict__ out,
                                 int n_nodes) {
    const int lane = threadIdx.x & 31;
    const int wid  = threadIdx.x >> 5;     // 0..7 -> column tile
    const int row0 = blockIdx.x * 16;
    const int col0 = wid * 16;
    const int half = lane >> 4;            // 0: K,K+1   1: K+2,K+3
    const int l15  = lane & 15;

    // clamp only matters for a (non-existent here) ragged final tile;
    // keeps EXEC all-1s through every WMMA.
    int arow_i = row0 + l15;
    if (arow_i >= n_nodes) arow_i = n_nodes - 1;
    const float* arow = agg + (size_t)arow_i * 128 + half * 2;
    const int    bcol = col0 + l15;

    v8f c = {};
#pragma unroll
    for (int k0 = 0; k0 < 128; k0 += 4) {
        v2f a, b;
        a.x = arow[k0 + 0];
        a.y = arow[k0 + 1];
        const int kb = k0 + half * 2;
        b.x = weight[(size_t)(kb + 0) * 128 + bcol];
        b.y = weight[(size_t)(kb + 1) * 128 + bcol];
        // 8 args: (neg_a, A, neg_b, B, c_mod, C, reuse_a, reuse_b)
        c = __builtin_amdgcn_wmma_f32_16x16x4_f32(
                false, a, false, b, (short)0, c, false, false);
    }

    const float bval = bias[bcol];
#pragma unroll
    for (int r = 0; r < 8; ++r) {
        const int m = row0 + r + half * 8;
        if (m < n_nodes) {
            out[(size_t)m * 128 + (col0 + l15)] = c[r] * norm[m] + bval;
        }
    }
}

// ---------------------------------------------------------------------------
// Launch
// ---------------------------------------------------------------------------
extern "C" void kernel_launch(void* const* d_in, const int* in_sizes, int n_in,
                              void* d_out, int out_size, void* d_ws, size_t ws_size,
                              hipStream_t stream) {
    const float* feat   = (const float*)d_in[0];
    const float* weight = (const float*)d_in[1];
    const float* bias   = (const float*)d_in[2];
    const int*   src    = (const int*)d_in[3];
    const int*   dst    = (const int*)d_in[4];
    float*       out    = (float*)d_out;

    const int n_nodes = in_sizes[0] / 128;   // 100000
    const int n_edges = in_sizes[3];         // 1600000

    float* deg = (float*)d_ws;               // n_nodes floats -> becomes norm
    float* agg = deg + n_nodes;              // n_nodes*128 floats (16B aligned)

    // 1) zero deg + agg  (n_nodes*129 floats, divisible by 4)
    {
        long n4 = ((long)n_nodes * 129) / 4;
        int blocks = 2048;
        zero_kernel<<<blocks, 256, 0, stream>>>((float4*)d_ws, n4);
    }
    // 2) in-degree
    deg_kernel<<<(n_edges + 255) / 256, 256, 0, stream>>>(dst, deg, n_edges);
    // 3) norm = clip(deg,1)^-0.5
    norm_kernel<<<(n_nodes + 255) / 256, 256, 0, stream>>>(deg, n_nodes);
    // 4) scatter-add: one wave per edge, 8 edges per 256-thread block
    {
        long threads = (long)n_edges * 32;
        int blocks = (int)((threads + 255) / 256);
        agg_kernel<<<blocks, 256, 0, stream>>>(feat, src, dst, deg, agg, n_edges);
    }
    // 5) fp32 WMMA GEMM with fused norm/bias epilogue
    gemm_wmma_kernel<<<(n_nodes + 15) / 16, 256, 0, stream>>>(
        agg, weight, bias, deg, out, n_nodes);
}